// MultiSiamLoss_9328668967444
// MI455X (gfx1250) — compile-verified
//
#include <hip/hip_runtime.h>

typedef __attribute__((ext_vector_type(2))) float v2f;
typedef __attribute__((ext_vector_type(8))) float v8f;
typedef int v4i __attribute__((vector_size(16)));   // matches builtin param pointee

#define MG        8                   // group size (reference M)
#define DD        256                 // embedding dim
#define RS        260                 // padded LDS row stride in dwords (4*r % 64 distinct for r<16)
#define GPB       8                   // groups handled per block
#define NROWS     ((GPB + 1) * MG)    // 72 rows staged (own 8 groups + next group for negatives)
#define EPS_DIST  1e-6f
#define EPS_NORM  1e-12f

#if __has_builtin(__builtin_amdgcn_global_load_async_to_lds_b128) && \
    __has_builtin(__builtin_amdgcn_s_wait_asynccnt)
#define USE_ASYNC_LDS 1
#else
#define USE_ASYNC_LDS 0
#endif

// One block = 128 threads = 4 waves; wave w computes 16 triplets via two
// f32 WMMA Gram accumulations (AP-diff and AN-diff); diagonal = squared dists.
__global__ __launch_bounds__(128) void siam_wmma_kernel(
    const float* __restrict__ x, float* __restrict__ partials, int B)
{
  extern __shared__ float lds[];          // NROWS * RS floats
  __shared__ float s_rnorm[NROWS];
  __shared__ float s_wsum[4];

  const int tid  = threadIdx.x;
  const int lane = tid & 31;
  const int wave = tid >> 5;
  const int g0   = blockIdx.x * GPB;      // first group owned by this block

  // ---- Stage 72 raw rows into padded LDS ----
  // CDNA5 path: GLOBAL_LOAD_ASYNC_TO_LDS_B128 (ASYNCcnt), no VGPR round-trip.
  for (int i = tid; i < NROWS * (DD / 4); i += 128) {
    const int r  = i >> 6;                // local row 0..71
    const int c4 = i & 63;                // float4 column 0..63
    const int lj = r >> 3, m = r & 7;     // local group, member
    const size_t grow = (size_t)(((g0 + lj) % B) * MG + m);
    const float4* src = ((const float4*)x) + grow * (DD / 4) + (size_t)c4;
    float* dst = &lds[r * RS + c4 * 4];   // 16B-aligned: (260*r + 4*c4) % 4 == 0
#if USE_ASYNC_LDS
    __builtin_amdgcn_global_load_async_to_lds_b128(
        (__attribute__((address_space(1))) v4i*)src,
        (__attribute__((address_space(3))) v4i*)dst,
        0, 0);
#else
    const float4 v = *src;
    dst[0] = v.x; dst[1] = v.y; dst[2] = v.z; dst[3] = v.w;
#endif
  }
#if USE_ASYNC_LDS
  __builtin_amdgcn_s_wait_asynccnt(0);    // my wave's async copies landed in LDS
#endif
  __syncthreads();                        // everyone's copies visible

  // ---- Row inverse norms: 18 rows per wave, wave-shuffle reduction ----
  for (int j = 0; j < NROWS / 4; ++j) {
    const int r = wave * (NROWS / 4) + j;
    const float* row = &lds[r * RS];
    float s = 0.0f;
#pragma unroll
    for (int k = 0; k < DD / 32; ++k) {
      const float v = row[lane + k * 32];
      s = fmaf(v, v, s);
    }
#pragma unroll
    for (int off = 16; off > 0; off >>= 1)
      s += __shfl_xor(s, off, 32);
    if (lane == 0) s_rnorm[r] = 1.0f / fmaxf(sqrtf(s), EPS_NORM);
  }
  __syncthreads();

  // ---- Gram accumulation: D += diff x diff^T over K=256 in chunks of 4 ----
  // A-layout (16x4 f32): lane<16 -> row=lane, k={0,1}; lane>=16 -> row=lane-16, k={2,3}.
  // B-layout (4x16 f32) is the column-major mirror, so SRC1==SRC0 gives A*A^T.
  const int gr   = lane & 15;             // gram row / triplet-in-wave
  const int koff = (lane >> 4) * 2;       // k sub-offset within chunk (0 or 2)
  const int lt   = wave * 16 + gr;        // local triplet 0..63
  const int lj   = lt >> 3, mm = lt & 7;
  const int la   = lj * MG + mm;                    // anchor local row
  const int lp   = lj * MG + ((mm + 1) & 7);        // positive: next member in group
  const int ln   = (lj + 1) * MG + mm;              // negative: same member, next group
  const float rna = s_rnorm[la], rnp = s_rnorm[lp], rnn = s_rnorm[ln];
  const float* arow = &lds[la * RS + koff];
  const float* prow = &lds[lp * RS + koff];
  const float* nrow = &lds[ln * RS + koff];

  v8f accAP = {0.f, 0.f, 0.f, 0.f, 0.f, 0.f, 0.f, 0.f};
  v8f accAN = {0.f, 0.f, 0.f, 0.f, 0.f, 0.f, 0.f, 0.f};

#pragma unroll 4
  for (int c = 0; c < DD; c += 4) {
    const float av0 = fmaf(arow[c],     rna, EPS_DIST);   // a*rna + eps
    const float av1 = fmaf(arow[c + 1], rna, EPS_DIST);
    v2f ap, an;
    ap.x = av0 - prow[c]     * rnp;
    ap.y = av1 - prow[c + 1] * rnp;
    an.x = av0 - nrow[c]     * rnn;
    an.y = av1 - nrow[c + 1] * rnn;
    accAP = __builtin_amdgcn_wmma_f32_16x16x4_f32(false, ap, false, ap,
                                                  (short)0, accAP, false, false);
    accAN = __builtin_amdgcn_wmma_f32_16x16x4_f32(false, an, false, an,
                                                  (short)0, accAN, false, false);
  }

  // ---- Extract diagonal: D[i][i] at lane i (i<8, c[i]) and lane i+16 (i>=8, c[i-8]) ----
  float contrib = 0.0f;
  if (lane < 8 || lane >= 24) {
    const int ci = lane & 7;
    float dap2 = accAP[0], dan2 = accAN[0];
#pragma unroll
    for (int k = 1; k < 8; ++k)
      if (ci == k) { dap2 = accAP[k]; dan2 = accAN[k]; }
    contrib = fmaxf(sqrtf(dap2) - sqrtf(dan2) + 1.0f, 0.0f);
  }
#pragma unroll
  for (int off = 16; off > 0; off >>= 1)
    contrib += __shfl_xor(contrib, off, 32);
  if (lane == 0) s_wsum[wave] = contrib;
  __syncthreads();
  if (tid == 0)
    partials[blockIdx.x] = (s_wsum[0] + s_wsum[1]) + (s_wsum[2] + s_wsum[3]);
}

// Deterministic final reduction of per-block partials -> mean.
__global__ __launch_bounds__(256) void siam_final_kernel(
    const float* __restrict__ partials, float* __restrict__ out,
    int n, float inv_count)
{
  __shared__ float sh[256];
  float s = 0.0f;
  for (int i = threadIdx.x; i < n; i += 256) s += partials[i];
  sh[threadIdx.x] = s;
  __syncthreads();
  for (int stride = 128; stride > 0; stride >>= 1) {
    if ((int)threadIdx.x < stride) sh[threadIdx.x] += sh[threadIdx.x + stride];
    __syncthreads();
  }
  if (threadIdx.x == 0) out[0] = sh[0] * inv_count;
}

extern "C" void kernel_launch(void* const* d_in, const int* in_sizes, int n_in,
                              void* d_out, int out_size, void* d_ws, size_t ws_size,
                              hipStream_t stream) {
  const float* x = (const float*)d_in[0];
  const int total = in_sizes[0];            // B*M*D
  const int B = total / (MG * DD);          // 16384 for the reference shape
  const int nblocks = B / GPB;              // 2048
  float* partials = (float*)d_ws;           // nblocks floats (8 KiB)

  const size_t shmem = (size_t)NROWS * RS * sizeof(float);  // ~73 KiB
  siam_wmma_kernel<<<nblocks, 128, shmem, stream>>>(x, partials, B);
  siam_final_kernel<<<1, 256, 0, stream>>>(partials, (float*)d_out, nblocks,
                                           1.0f / (float)(B * MG));
}